// CollectNeighbourAverageAndMax_38285338476787
// MI455X (gfx1250) — compile-verified
//
#include <hip/hip_runtime.h>

// CollectNeighbourAverageAndMax for MI455X (gfx1250, wave32).
// V=100000 vertices, K=32 neighbors, F=64 f32 features.
// One wave per vertex: 32 lanes x float2 = one full 256B feature row per load.
// Per-neighbor row base is pinned into SGPRs so the gathers use the
// saddr+voffset form (scalar address math, zero address VGPRs).

typedef float v2f __attribute__((ext_vector_type(2)));

#define KNN 32
#define FEAT 64
#define WAVES_PER_BLOCK 8   // 256 threads = 8 wave32

__global__ __launch_bounds__(256) void
CollectNeighbourAverageAndMax_kernel(const float* __restrict__ x,
                                     const int* __restrict__ idxs,
                                     float* __restrict__ out,
                                     int V)
{
    const int lane = threadIdx.x & 31;
    const int v    = blockIdx.x * WAVES_PER_BLOCK + (threadIdx.x >> 5);
    if (v >= V) return;   // wave-uniform exit

    // Each lane picks up one of this vertex's 32 neighbor indices.
    // Read exactly once -> non-temporal, keep L2 filled with the 25.6MB table.
    int myIdx = __builtin_nontemporal_load(&idxs[v * KNN + lane]);

    const int laneByte = lane << 3;            // loop-invariant VGPR offset

    v2f s = {0.0f, 0.0f};
    v2f m = {-__builtin_inff(), -__builtin_inff()};

#pragma unroll
    for (int k = 0; k < KNN; ++k) {
        // Broadcast neighbor k's index into an SGPR (v_readlane_b32, imm lane).
        int j = __builtin_amdgcn_readlane(myIdx, k);

        // Uniform row base, pinned into an SGPR pair: blocks the compiler from
        // reassociating to a per-lane 64-bit VGPR address, so the load selects
        // the saddr form: global_load_b64 vdst, v_laneByte, s[base:base+1].
        const char* base = (const char*)x + (long long)j * (FEAT * sizeof(float));
        asm("" : "+s"(base));

        v2f g = *(const v2f*)(base + laneByte);  // coalesced 256B row
        s.x += g.x;
        s.y += g.y;
        m.x = __builtin_fmaxf(m.x, g.x);
        m.y = __builtin_fmaxf(m.y, g.y);
    }

    const float inv = 1.0f / (float)KNN;
    v2f mean = {s.x * inv, s.y * inv};

    // Output is write-once -> NT stores so the 51MB output stream doesn't
    // evict the L2-resident gather table.
    v2f* outv = reinterpret_cast<v2f*>(out + (size_t)v * (2 * FEAT));
    __builtin_nontemporal_store(mean, &outv[lane]);            // features [0,64)
    __builtin_nontemporal_store(m,    &outv[FEAT / 2 + lane]); // features [64,128)
}

extern "C" void kernel_launch(void* const* d_in, const int* in_sizes, int n_in,
                              void* d_out, int out_size, void* d_ws, size_t ws_size,
                              hipStream_t stream)
{
    const float* x    = (const float*)d_in[0];
    const int*   idxs = (const int*)d_in[1];
    float*       out  = (float*)d_out;

    const int V = in_sizes[0] / FEAT;                 // 100000
    const int blocks = (V + WAVES_PER_BLOCK - 1) / WAVES_PER_BLOCK;

    CollectNeighbourAverageAndMax_kernel<<<blocks, 256, 0, stream>>>(x, idxs, out, V);
}